// Lstm_59562606461331
// MI455X (gfx1250) — compile-verified
//
#include <hip/hip_runtime.h>

// ---------------------------------------------------------------------------
// ConvLSTM2D (B=2,T=16,H=W=64,C=F=64,K=3) + inference BatchNorm for gfx1250.
// Convs are implicit GEMM on V_WMMA_F32_16X16X4_F32 (f32 in, f32 acc).
// Input conv: 32-pixel x 256-ch block tiles (2 A-frags/wave -> 8 wmma per
// 4 B loads, 16 flops/L2-byte). Recurrent conv: 16-pixel tiles for grid fill.
// ---------------------------------------------------------------------------

typedef __attribute__((ext_vector_type(2))) float v2f;
typedef __attribute__((ext_vector_type(8))) float v8f;

#define HW            64
#define CIN           64
#define NF            64
#define NGATES        256        // 4*NF
#define KQ            144        // 576 / 4 k-quads
#define PATCH_STRIDE  68         // 64 ch + 4 pad floats -> conflict-free 8B LDS reads
#define GATE_STRIDE   66         // 64 + 2 pad -> conflict-free 4B LDS writes

// --- prep: swizzle weights (k,n) f32 -> per-lane WMMA B-fragment float2 pairs.
// Wr[(q*2+h)*256 + n] = { W[(4q+2h)*256+n], W[(4q+2h+1)*256+n] }
__global__ void reorder_weights(const float* __restrict__ src, float2* __restrict__ dst) {
    int i = blockIdx.x * blockDim.x + threadIdx.x;        // KQ*2*256 = 73728
    if (i >= KQ * 2 * NGATES) return;
    int n  = i & (NGATES - 1);
    int hh = (i >> 8) & 1;
    int q  = i >> 9;
    int k0 = 4 * q + 2 * hh;
    dst[i] = make_float2(src[k0 * NGATES + n], src[(k0 + 1) * NGATES + n]);
}

__global__ void zero_f32(float* __restrict__ p, int n) {
    int i = blockIdx.x * blockDim.x + threadIdx.x;
    if (i < n) p[i] = 0.0f;
}

// --- shared implicit-GEMM conv tile: MP*16 pixels (row y, cols x0..) x 256 ch.
// Block = 128 threads = 4 waves; wave w owns output channels [w*64, w*64+64).
// acc[pair][sub] accumulates pixels [pair*16, pair*16+16) x ch [w*64+sub*16, +16).
template <int MP>
__device__ __forceinline__ void conv_tile_wmma(const float* __restrict__ src,   // image base (64,64,CIN) NHWC
                                               const float2* __restrict__ wr,   // reordered weights
                                               float* __restrict__ patch,       // LDS 3*(16*MP+2)*PATCH_STRIDE
                                               int y, int x0, v8f acc[MP][4]) {
    constexpr int PC = 16 * MP + 2;                       // patch columns incl. halo
    const int tid = threadIdx.x;
    // stage 3 x PC x 64 halo patch with zero padding
    for (int e = tid; e < 3 * PC * CIN; e += 128) {
        int ch  = e & (CIN - 1);
        int rc  = e >> 6;                 // r*PC + col
        int col = rc % PC;
        int r   = rc / PC;
        int yy  = y + r - 1;
        int xx  = x0 + col - 1;
        float v = 0.0f;
        if (yy >= 0 && yy < HW && xx >= 0 && xx < HW)
            v = src[((yy * HW) + xx) * CIN + ch];
        patch[(r * PC + col) * PATCH_STRIDE + ch] = v;
    }
    __syncthreads();

    const int lane = tid & 31;
    const int wv   = tid >> 5;            // wave id 0..3 -> channel group / gate
    const int m    = lane & 15;           // A: row M / B,C: col N
    const int hh   = lane >> 4;           // k-half select
    const int nb   = wv * 64 + m;         // base output channel of this lane

    for (int ky = 0; ky < 3; ++ky) {
        for (int kx = 0; kx < 3; ++kx) {
            const float* prow[MP];
#pragma unroll
            for (int pr = 0; pr < MP; ++pr)
                prow[pr] = &patch[(ky * PC + pr * 16 + m + kx) * PATCH_STRIDE + 2 * hh];
            const float2* wq = wr + (size_t)((ky * 3 + kx) * 16) * 2 * NGATES;
#pragma unroll
            for (int cq = 0; cq < 16; ++cq) {
                v2f a[MP];
#pragma unroll
                for (int pr = 0; pr < MP; ++pr) {
                    float2 av = *(const float2*)(prow[pr] + 4 * cq);   // ds_load_b64
                    a[pr].x = av.x; a[pr].y = av.y;
                }
                const float2* wp = wq + (size_t)(cq * 2 + hh) * NGATES + nb;
#pragma unroll
                for (int sub = 0; sub < 4; ++sub) {
                    float2 bw = wp[sub * 16];
                    v2f b; b.x = bw.x; b.y = bw.y;
#pragma unroll
                    for (int pr = 0; pr < MP; ++pr)
                        acc[pr][sub] = __builtin_amdgcn_wmma_f32_16x16x4_f32(
                            false, a[pr], false, b, (short)0, acc[pr][sub], false, false);
                }
            }
        }
    }
}

// --- kernel 1: xg = conv(x, Wx) + b for all 32 frames. grid = 32*64*2 blocks,
// each block computes a 32-pixel row segment x 256 gate channels.
__global__ __launch_bounds__(128) void conv_in_kernel(const float* __restrict__ x,
                                                      const float2* __restrict__ wxr,
                                                      const float* __restrict__ bias,
                                                      float* __restrict__ xg) {
    __shared__ float patch[3 * 34 * PATCH_STRIDE];        // 27.7 KB
    int blk = blockIdx.x;
    int xt  = blk & 1;
    int y   = (blk >> 1) & 63;
    int img = blk >> 7;                  // 0..31
    int x0  = xt * 32;

    v8f zero = {0.f, 0.f, 0.f, 0.f, 0.f, 0.f, 0.f, 0.f};
    v8f acc[2][4] = {{zero, zero, zero, zero}, {zero, zero, zero, zero}};
    conv_tile_wmma<2>(x + (size_t)img * HW * HW * CIN, wxr, patch, y, x0, acc);

    const int lane = threadIdx.x & 31;
    const int wv   = threadIdx.x >> 5;
    const int m    = lane & 15;
    const int hh   = lane >> 4;
    float* outb = xg + ((size_t)img * HW * HW + y * HW + x0) * NGATES;
#pragma unroll
    for (int pr = 0; pr < 2; ++pr) {
#pragma unroll
        for (int sub = 0; sub < 4; ++sub) {
            int n    = wv * 64 + sub * 16 + m;
            float bb = bias[n];
#pragma unroll
            for (int r = 0; r < 8; ++r) {
                int p = pr * 16 + r + 8 * hh;            // pixel (C-matrix layout)
                outb[(size_t)p * NGATES + n] = acc[pr][sub][r] + bb;
            }
        }
    }
}

// --- kernel 2: one LSTM timestep. grid = 2*64*4 blocks (16-pixel tiles: keeps
// the serial per-timestep chain wide enough to fill the machine).
__global__ __launch_bounds__(128) void step_kernel(const float* __restrict__ hprev,
                                                   const float2* __restrict__ whr,
                                                   const float* __restrict__ xg,
                                                   float* __restrict__ hnext,
                                                   float* __restrict__ cbuf,
                                                   const float* __restrict__ gamma,
                                                   const float* __restrict__ beta,
                                                   const float* __restrict__ mean,
                                                   const float* __restrict__ var,
                                                   float* __restrict__ out,
                                                   int t, int has_h) {
    __shared__ float patch[3 * 18 * PATCH_STRIDE];
    __shared__ float gates[4 * 16 * GATE_STRIDE];

    int blk = blockIdx.x;
    int xt  = blk & 3;
    int y   = (blk >> 2) & 63;
    int b   = blk >> 8;                  // batch 0..1
    int x0  = xt * 16;

    v8f zero = {0.f, 0.f, 0.f, 0.f, 0.f, 0.f, 0.f, 0.f};
    v8f acc[1][4] = {{zero, zero, zero, zero}};
    if (has_h)                           // uniform branch; h0 == 0 so skip conv at t=0
        conv_tile_wmma<1>(hprev + (size_t)b * HW * HW * NF, whr, patch, y, x0, acc);

    const int lane = threadIdx.x & 31;
    const int wv   = threadIdx.x >> 5;   // wave wv holds gate wv (64 ch each)
    const int m    = lane & 15;
    const int hh   = lane >> 4;
#pragma unroll
    for (int sub = 0; sub < 4; ++sub) {
#pragma unroll
        for (int r = 0; r < 8; ++r) {
            int p = r + 8 * hh;
            gates[(wv * 16 + p) * GATE_STRIDE + sub * 16 + m] = acc[0][sub][r];
        }
    }
    __syncthreads();

    // fused: z = conv + xg, gate nonlinearities, c/h update, BatchNorm, output
    const float* xgp = xg + ((size_t)(b * 16 + t) * HW * HW + y * HW + x0) * NGATES;
    float* cb = cbuf  + ((size_t)b * HW * HW + y * HW + x0) * NF;
    float* hb = hnext + ((size_t)b * HW * HW + y * HW + x0) * NF;
    float* ob = out   + ((size_t)(b * 16 + t) * HW * HW + y * HW + x0) * NF;

    for (int e = threadIdx.x; e < 16 * NF; e += 128) {
        int j = e & (NF - 1);
        int p = e >> 6;
        float zi = gates[(0 * 16 + p) * GATE_STRIDE + j] + xgp[p * NGATES + 0 * 64 + j];
        float zf = gates[(1 * 16 + p) * GATE_STRIDE + j] + xgp[p * NGATES + 1 * 64 + j];
        float zg = gates[(2 * 16 + p) * GATE_STRIDE + j] + xgp[p * NGATES + 2 * 64 + j];
        float zo = gates[(3 * 16 + p) * GATE_STRIDE + j] + xgp[p * NGATES + 3 * 64 + j];
        float ig = fminf(fmaxf(0.2f * zi + 0.5f, 0.0f), 1.0f);
        float fg = fminf(fmaxf(0.2f * zf + 0.5f, 0.0f), 1.0f);
        float gg = tanhf(zg);
        float og = fminf(fmaxf(0.2f * zo + 0.5f, 0.0f), 1.0f);
        float cold = cb[p * NF + j];
        float cnew = fg * cold + ig * gg;
        float hnew = og * tanhf(cnew);
        cb[p * NF + j] = cnew;
        hb[p * NF + j] = hnew;
        float inv = gamma[j] * rsqrtf(var[j] + 1e-3f);
        ob[p * NF + j] = hnew * inv + (beta[j] - mean[j] * inv);
    }
}

extern "C" void kernel_launch(void* const* d_in, const int* in_sizes, int n_in,
                              void* d_out, int out_size, void* d_ws, size_t ws_size,
                              hipStream_t stream) {
    const float* x     = (const float*)d_in[0];
    const float* Wx    = (const float*)d_in[1];
    const float* Wh    = (const float*)d_in[2];
    const float* bias  = (const float*)d_in[3];
    const float* gamma = (const float*)d_in[4];
    const float* beta  = (const float*)d_in[5];
    const float* mean  = (const float*)d_in[6];
    const float* var   = (const float*)d_in[7];
    float* out = (float*)d_out;

    // workspace carve-up (all re-initialized every call; nothing persists)
    char* ws = (char*)d_ws;
    float2* wxr = (float2*)ws;  ws += (size_t)KQ * 2 * NGATES * sizeof(float2);   // 576 KB
    float2* whr = (float2*)ws;  ws += (size_t)KQ * 2 * NGATES * sizeof(float2);   // 576 KB
    float*  xg  = (float*)ws;   ws += (size_t)32 * HW * HW * NGATES * sizeof(float); // 128 MB
    float*  h0  = (float*)ws;   ws += (size_t)2 * HW * HW * NF * sizeof(float);   // 2 MB
    float*  h1  = (float*)ws;   ws += (size_t)2 * HW * HW * NF * sizeof(float);   // 2 MB
    float*  cbf = (float*)ws;                                                     // 2 MB

    reorder_weights<<<(KQ * 2 * NGATES + 255) / 256, 256, 0, stream>>>(Wx, wxr);
    reorder_weights<<<(KQ * 2 * NGATES + 255) / 256, 256, 0, stream>>>(Wh, whr);

    const int csz = 2 * HW * HW * NF;
    zero_f32<<<(csz + 255) / 256, 256, 0, stream>>>(cbf, csz);

    conv_in_kernel<<<32 * HW * 2, 128, 0, stream>>>(x, wxr, bias, xg);

    float* hb[2] = {h0, h1};
    for (int t = 0; t < 16; ++t) {
        step_kernel<<<2 * HW * 4, 128, 0, stream>>>(hb[t & 1], whr, xg,
                                                    hb[(t + 1) & 1], cbf,
                                                    gamma, beta, mean, var, out,
                                                    t, t > 0 ? 1 : 0);
    }
}